// MultiScaleWaveletBlock_89584427860566
// MI455X (gfx1250) — compile-verified
//
#include <hip/hip_runtime.h>

// MultiScaleWaveletBlock fused kernel for MI455X (gfx1250).
// One pass: global -> LDS tile (async DMA path) -> 3-level Haar + linear
// upsample of 4 bands -> coalesced float4 stores. Memory-bound: ~320MB moved,
// ideal ~14us @ 23.3 TB/s HBM.

#define B_DIM   16
#define T_DIM   2048
#define C_DIM   512

#define T_TILE  128
#define C_TILE  64
#define HALO_LO 8
#define HALO_HI 8
#define ROWS    (T_TILE + HALO_LO + HALO_HI)   // 144
#define NTHREADS 512
#define ROWSTEP (NTHREADS / C_TILE)            // 8

#define HC 0.70710678118654752440f             // 1/sqrt(2)

#if defined(__has_builtin)
#if __has_builtin(__builtin_amdgcn_global_load_async_to_lds_b32)
#define USE_ASYNC_LDS 1
#endif
#if __has_builtin(__builtin_amdgcn_s_wait_asynccnt)
#define HAVE_WAIT_ASYNC 1
#endif
#endif

__global__ __launch_bounds__(NTHREADS) void
MultiScaleWaveletBlock_89584427860566_kernel(const float* __restrict__ x,
                                             float* __restrict__ out)
{
    __shared__ float smem[ROWS * C_TILE];

    const int tid   = threadIdx.x;
    const int cL    = tid & (C_TILE - 1);
    const int C0    = blockIdx.x * C_TILE;   // 8 c-tiles
    const int T0    = blockIdx.y * T_TILE;   // 16 t-tiles
    const int b     = blockIdx.z;            // 16 batches
    const int tBase = T0 - HALO_LO;

    // ---------------- Stage tile (ROWS x C_TILE) into LDS ----------------
    for (int r = (tid >> 6); r < ROWS; r += ROWSTEP) {
        int ti = tBase + r;
        ti = ti < 0 ? 0 : (ti > (T_DIM - 1) ? (T_DIM - 1) : ti);  // clamped halo rows never read
        const float* src = x + ((size_t)b * T_DIM + ti) * C_DIM + C0 + cL;
        float*       dst = &smem[r * C_TILE + cL];
#if defined(USE_ASYNC_LDS)
        __builtin_amdgcn_global_load_async_to_lds_b32((int*)src, (int*)dst, 0, 0);
#else
        *dst = *src;
#endif
    }
#if defined(USE_ASYNC_LDS)
#if defined(HAVE_WAIT_ASYNC)
    __builtin_amdgcn_s_wait_asynccnt(0);
#else
    asm volatile("s_wait_asynccnt 0" ::: "memory");
#endif
#endif
    __syncthreads();

#define LD(ti) smem[((ti) - tBase) * C_TILE + cL]

    // ---------------- Compute 4 bands per (t, c) ----------------
    const int tLane = tid >> 6;  // 0..7
    for (int it = 0; it < T_TILE / ROWSTEP; ++it) {
        const int   t  = T0 + tLane + it * ROWSTEP;
        const float tf = (float)t;
        float o0, o1, o2, o3;

        // Band 0: interp of level-1 detail (L = 1024, scale = 0.5)
        {
            float src = fmaxf((tf + 0.5f) * 0.5f - 0.5f, 0.0f);
            int x0 = (int)src;  x0 = x0 > 1023 ? 1023 : x0;
            int x1 = x0 + 1;    x1 = x1 > 1023 ? 1023 : x1;
            float w = src - (float)x0;
            float v0 = (LD(2 * x0) - LD(2 * x0 + 1)) * HC;
            float v1 = (LD(2 * x1) - LD(2 * x1 + 1)) * HC;
            o0 = v0 * (1.0f - w) + v1 * w;
        }

        // Band 1: interp of level-2 detail (L = 512, scale = 0.25)
        {
            float src = fmaxf((tf + 0.5f) * 0.25f - 0.5f, 0.0f);
            int x0 = (int)src;  x0 = x0 > 511 ? 511 : x0;
            int x1 = x0 + 1;    x1 = x1 > 511 ? 511 : x1;
            float w = src - (float)x0;
            int j = 4 * x0;
            float a  = (LD(j)     + LD(j + 1)) * HC;
            float bb = (LD(j + 2) + LD(j + 3)) * HC;
            float v0 = (a - bb) * HC;
            j = 4 * x1;
            a  = (LD(j)     + LD(j + 1)) * HC;
            bb = (LD(j + 2) + LD(j + 3)) * HC;
            float v1 = (a - bb) * HC;
            o1 = v0 * (1.0f - w) + v1 * w;
        }

        // Bands 2 & 3: level-3 detail + approximation (both L = 256, scale = 0.125)
        {
            float src = fmaxf((tf + 0.5f) * 0.125f - 0.5f, 0.0f);
            int x0 = (int)src;  x0 = x0 > 255 ? 255 : x0;
            int x1 = x0 + 1;    x1 = x1 > 255 ? 255 : x1;
            float w = src - (float)x0;
            float d3v0, a3v0, d3v1, a3v1;
            {
                int j = 8 * x0;
                float a10 = (LD(j)     + LD(j + 1)) * HC;
                float a11 = (LD(j + 2) + LD(j + 3)) * HC;
                float a12 = (LD(j + 4) + LD(j + 5)) * HC;
                float a13 = (LD(j + 6) + LD(j + 7)) * HC;
                float a20 = (a10 + a11) * HC;
                float a21 = (a12 + a13) * HC;
                d3v0 = (a20 - a21) * HC;
                a3v0 = (a20 + a21) * HC;
            }
            {
                int j = 8 * x1;
                float a10 = (LD(j)     + LD(j + 1)) * HC;
                float a11 = (LD(j + 2) + LD(j + 3)) * HC;
                float a12 = (LD(j + 4) + LD(j + 5)) * HC;
                float a13 = (LD(j + 6) + LD(j + 7)) * HC;
                float a20 = (a10 + a11) * HC;
                float a21 = (a12 + a13) * HC;
                d3v1 = (a20 - a21) * HC;
                a3v1 = (a20 + a21) * HC;
            }
            o2 = d3v0 * (1.0f - w) + d3v1 * w;
            o3 = a3v0 * (1.0f - w) + a3v1 * w;
        }

        // Coalesced 16B store: out[b][t][c][0..3]
        float4 o = make_float4(o0, o1, o2, o3);
        *reinterpret_cast<float4*>(
            out + (((size_t)b * T_DIM + t) * C_DIM + C0 + cL) * 4) = o;
    }
#undef LD
}

extern "C" void kernel_launch(void* const* d_in, const int* in_sizes, int n_in,
                              void* d_out, int out_size, void* d_ws, size_t ws_size,
                              hipStream_t stream) {
    (void)in_sizes; (void)n_in; (void)d_ws; (void)ws_size; (void)out_size;
    const float* x  = (const float*)d_in[0];
    float* out      = (float*)d_out;
    dim3 grid(C_DIM / C_TILE, T_DIM / T_TILE, B_DIM);  // (8, 16, 16)
    MultiScaleWaveletBlock_89584427860566_kernel<<<grid, NTHREADS, 0, stream>>>(x, out);
}